// CBAM_attention_43430709297173
// MI455X (gfx1250) — compile-verified
//
#include <hip/hip_runtime.h>
#include <hip/hip_bf16.h>
#include <math.h>

typedef __attribute__((ext_vector_type(2))) float v2f;
typedef __attribute__((ext_vector_type(8))) float v8f;

#define EPSF 1e-12f

__device__ __forceinline__ float sigmoidf(float v) { return 1.0f / (1.0f + expf(-v)); }

// ---------------------------------------------------------------------------
// Kernel 1: per-(b,c) mean & max over H*W=4096.  red layout: avg[0..16383],
// max[16384..32767]  ==> stacked A matrix [64][512] for the WMMA MLP.
// ---------------------------------------------------------------------------
__global__ __launch_bounds__(256) void reduce_hw_kernel(
    const float* __restrict__ x, float* __restrict__ avg, float* __restrict__ mx) {
  const int bc = blockIdx.x;                     // 0..16383  (b*512+c)
  const int tid = threadIdx.x;
  const float4* p4 = (const float4*)(x + (size_t)bc * 4096);
  float s = 0.0f, m = -INFINITY;
  #pragma unroll
  for (int i = tid; i < 1024; i += 256) {
    float4 v = p4[i];
    s += v.x + v.y + v.z + v.w;
    m = fmaxf(m, fmaxf(fmaxf(v.x, v.y), fmaxf(v.z, v.w)));
  }
  __shared__ float ss[256], sm[256];
  ss[tid] = s; sm[tid] = m;
  __syncthreads();
  for (int o = 128; o > 0; o >>= 1) {
    if (tid < o) { ss[tid] += ss[tid + o]; sm[tid] = fmaxf(sm[tid], sm[tid + o]); }
    __syncthreads();
  }
  if (tid == 0) { avg[bc] = ss[0] * (1.0f / 4096.0f); mx[bc] = sm[0]; }
}

// ---------------------------------------------------------------------------
// Kernel 2: spectral norms (1 power iteration, eval mode) -> inv_sv[3]
// ---------------------------------------------------------------------------
__device__ float block_reduce_sum(float v, float* red) {
  const int tid = threadIdx.x;
  __syncthreads();
  red[tid] = v;
  __syncthreads();
  for (int o = 128; o > 0; o >>= 1) {
    if (tid < o) red[tid] += red[tid + o];
    __syncthreads();
  }
  float r = red[0];
  __syncthreads();
  return r;
}

__global__ __launch_bounds__(256) void sn_kernel(
    const float* __restrict__ w1, const float* __restrict__ u1,
    const float* __restrict__ w2, const float* __restrict__ u2,
    const float* __restrict__ w3, const float* __restrict__ u3,
    float* __restrict__ invsv) {
  const int tid = threadIdx.x;
  __shared__ float red[256];
  __shared__ float vsh[512];
  __shared__ float v2sh[64];

  // ---- sv1: Wm = w1 [64,512], u1 [1,64] ----
  for (int j = tid; j < 512; j += 256) {
    float acc = 0.0f;
    for (int i = 0; i < 64; ++i) acc += u1[i] * w1[i * 512 + j];
    vsh[j] = acc;
  }
  __syncthreads();
  float part = 0.0f;
  for (int j = tid; j < 512; j += 256) part += vsh[j] * vsh[j];
  float n2 = block_reduce_sum(part, red);
  float vs = 1.0f / fmaxf(sqrtf(n2), EPSF);
  float tpart = 0.0f;
  if (tid < 64) {
    float acc = 0.0f;
    for (int j = 0; j < 512; ++j) acc += vsh[j] * w1[tid * 512 + j];
    acc *= vs;
    tpart = acc * acc;
  }
  float t2 = block_reduce_sum(tpart, red);
  if (tid == 0) {
    float sv = t2 / fmaxf(sqrtf(t2), EPSF);
    invsv[0] = 1.0f / sv;
  }

  // ---- sv2: Wm = w2 [512,64], u2 [1,512] ----
  if (tid < 64) {
    float acc = 0.0f;
    for (int i = 0; i < 512; ++i) acc += u2[i] * w2[i * 64 + tid];
    v2sh[tid] = acc;
  }
  __syncthreads();
  part = (tid < 64) ? v2sh[tid] * v2sh[tid] : 0.0f;
  n2 = block_reduce_sum(part, red);
  vs = 1.0f / fmaxf(sqrtf(n2), EPSF);
  tpart = 0.0f;
  for (int i = tid; i < 512; i += 256) {
    float acc = 0.0f;
    for (int j = 0; j < 64; ++j) acc += v2sh[j] * w2[i * 64 + j];
    acc *= vs;
    tpart += acc * acc;
  }
  t2 = block_reduce_sum(tpart, red);
  if (tid == 0) {
    float sv = t2 / fmaxf(sqrtf(t2), EPSF);
    invsv[1] = 1.0f / sv;
  }

  // ---- sv3: Wm = w3 flat [1,18], u3 scalar ----
  if (tid == 0) {
    float u = u3[0];
    float nv = 0.0f;
    for (int j = 0; j < 18; ++j) { float vj = u * w3[j]; nv += vj * vj; }
    float vs3 = 1.0f / fmaxf(sqrtf(nv), EPSF);
    float t = 0.0f;
    for (int j = 0; j < 18; ++j) t += (u * w3[j] * vs3) * w3[j];
    float sv = (t * t) / fmaxf(fabsf(t), EPSF);
    invsv[2] = 1.0f / sv;
  }
}

// ---------------------------------------------------------------------------
// Kernel 3: MLP via V_WMMA_F32_16X16X4_F32.
// A = stacked [64,512] (rows 0..31 = avg, 32..63 = max).
// GEMM1: D1 = A x W1^T  (64x64x512), h = relu(inv1*D1 + b1)    -> LDS
// GEMM2: D2 = h x W2^T  (64x512x64)                            -> g2 (global)
// s[b][c] = sigmoid(inv2*(D2[b][c] + D2[b+32][c]) + 2*b2[c])
// ---------------------------------------------------------------------------
__global__ __launch_bounds__(256) void mlp_wmma_kernel(
    const float* __restrict__ red,   // [64*512] stacked A
    const float* __restrict__ w1,    // [64*512]
    const float* __restrict__ b1,    // [64]
    const float* __restrict__ w2,    // [512*64]
    const float* __restrict__ b2,    // [512]
    const float* __restrict__ invsv, // [3]
    float* __restrict__ g2,          // [64*512] scratch
    float* __restrict__ s_out) {     // [32*512]
  __shared__ float h_lds[64 * 64];

  const int tid   = threadIdx.x;
  const int wave  = tid >> 5;        // 0..7
  const int lane  = tid & 31;
  const int lr    = lane & 15;       // row-in-tile / col-in-tile
  const int khalf = lane >> 4;       // 0: K pair {0,1}, 1: K pair {2,3}
  const float inv1 = invsv[0];
  const float inv2 = invsv[1];

  // ---------------- GEMM1: M=64, N=64, K=512 (4x4 tiles of 16x16) ----------
  for (int t = wave; t < 16; t += 8) {
    const int mt = t >> 2, nt = t & 3;
    const int m0 = mt * 16 + lr;      // A row for this lane
    const int n0 = nt * 16 + lr;      // B col for this lane
    v8f acc = {};
    for (int k0 = 0; k0 < 512; k0 += 4) {
      v2f a = *(const v2f*)(red + m0 * 512 + k0 + 2 * khalf);
      v2f b = *(const v2f*)(w1  + n0 * 512 + k0 + 2 * khalf);
      acc = __builtin_amdgcn_wmma_f32_16x16x4_f32(
          /*neg_a=*/false, a, /*neg_b=*/false, b,
          /*c_mod=*/(short)0, acc, /*reuse_a=*/false, /*reuse_b=*/false);
    }
    // D tile layout: VGPR r -> (M = mt*16 + r + khalf*8, N = nt*16 + lr)
    const float bias = b1[nt * 16 + lr];
    #pragma unroll
    for (int r = 0; r < 8; ++r) {
      const int m = mt * 16 + r + khalf * 8;
      h_lds[m * 64 + nt * 16 + lr] = fmaxf(inv1 * acc[r] + bias, 0.0f);
    }
  }
  __syncthreads();

  // ---------------- GEMM2: M=64, N=512, K=64 (4x32 tiles) ------------------
  for (int t = wave; t < 128; t += 8) {
    const int mt = t >> 5, nt = t & 31;
    const int m0 = mt * 16 + lr;
    const int n0 = nt * 16 + lr;
    v8f acc = {};
    for (int k0 = 0; k0 < 64; k0 += 4) {
      v2f a = *(const v2f*)(h_lds + m0 * 64 + k0 + 2 * khalf);
      v2f b = *(const v2f*)(w2    + n0 * 64 + k0 + 2 * khalf);
      acc = __builtin_amdgcn_wmma_f32_16x16x4_f32(
          false, a, false, b, (short)0, acc, false, false);
    }
    #pragma unroll
    for (int r = 0; r < 8; ++r) {
      const int m = mt * 16 + r + khalf * 8;
      g2[m * 512 + nt * 16 + lr] = acc[r];
    }
  }
  __syncthreads();

  // ---------------- combine: s = sigmoid(inv2*(Da+Dm) + 2*b2) --------------
  for (int i = tid; i < 32 * 512; i += 256) {
    const int b = i >> 9, c = i & 511;
    float v = inv2 * (g2[b * 512 + c] + g2[(b + 32) * 512 + c]) + 2.0f * b2[c];
    s_out[i] = sigmoidf(v);
  }
}

// ---------------------------------------------------------------------------
// Kernel 4: channel mean & max of x*s over C=512  -> ca, cm  [B,4096]
// grid (16, 32): block handles 256 spatial positions of one batch.
// ---------------------------------------------------------------------------
__global__ __launch_bounds__(256) void channel_reduce_kernel(
    const float* __restrict__ x, const float* __restrict__ s,
    float* __restrict__ ca, float* __restrict__ cm) {
  const int b = blockIdx.y;
  const int p = blockIdx.x * 256 + threadIdx.x;   // 0..4095
  __shared__ float s_sh[512];
  for (int c = threadIdx.x; c < 512; c += 256) s_sh[c] = s[b * 512 + c];
  __syncthreads();
  const float* xb = x + (size_t)b * 512 * 4096;
  float acc = 0.0f, mxv = -INFINITY;
  #pragma unroll 4
  for (int c = 0; c < 512; ++c) {
    float v = xb[(size_t)c * 4096 + p] * s_sh[c];
    acc += v;
    mxv = fmaxf(mxv, v);
  }
  ca[b * 4096 + p] = acc * (1.0f / 512.0f);
  cm[b * 4096 + p] = mxv;
}

// ---------------------------------------------------------------------------
// Kernel 5: 3x3 conv (2-in, 1-out, zero pad) + bias + sigmoid -> sattn
// ---------------------------------------------------------------------------
__global__ __launch_bounds__(256) void spatial_conv_kernel(
    const float* __restrict__ ca, const float* __restrict__ cm,
    const float* __restrict__ w3, const float* __restrict__ b3,
    const float* __restrict__ invsv, float* __restrict__ sattn) {
  const int b = blockIdx.y;
  const int p = blockIdx.x * 256 + threadIdx.x;   // 0..4095
  const int h = p >> 6, w = p & 63;
  const float inv3 = invsv[2];
  float acc = b3[0];
  #pragma unroll
  for (int kh = 0; kh < 3; ++kh) {
    const int hh = h + kh - 1;
    if (hh < 0 || hh > 63) continue;
    #pragma unroll
    for (int kw = 0; kw < 3; ++kw) {
      const int ww = w + kw - 1;
      if (ww < 0 || ww > 63) continue;
      const int q = b * 4096 + hh * 64 + ww;
      acc += ca[q] * (w3[kh * 3 + kw] * inv3);
      acc += cm[q] * (w3[9 + kh * 3 + kw] * inv3);
    }
  }
  sattn[b * 4096 + p] = sigmoidf(acc);
}

// ---------------------------------------------------------------------------
// Kernel 6: out = x * s[b,c] * sattn[b,p]   (float4)
// ---------------------------------------------------------------------------
__global__ __launch_bounds__(256) void final_mul_kernel(
    const float* __restrict__ x, const float* __restrict__ s,
    const float* __restrict__ sattn, float* __restrict__ out) {
  const size_t i4 = (size_t)blockIdx.x * 256 + threadIdx.x;  // < 16777216
  const int b = (int)(i4 >> 19);          // 512*4096/4 = 2^19 float4 per batch
  const int c = (int)(i4 >> 10) & 511;    // 4096/4 = 1024 float4 per channel
  const int p = ((int)i4 & 1023) * 4;
  float4 v = ((const float4*)x)[i4];
  const float sv = s[b * 512 + c];
  float4 a = *(const float4*)(sattn + b * 4096 + p);
  float4 o;
  o.x = v.x * sv * a.x;
  o.y = v.y * sv * a.y;
  o.z = v.z * sv * a.z;
  o.w = v.w * sv * a.w;
  ((float4*)out)[i4] = o;
}

// ---------------------------------------------------------------------------
extern "C" void kernel_launch(void* const* d_in, const int* in_sizes, int n_in,
                              void* d_out, int out_size, void* d_ws, size_t ws_size,
                              hipStream_t stream) {
  const float* x  = (const float*)d_in[0];
  const float* w1 = (const float*)d_in[1];
  const float* b1 = (const float*)d_in[2];
  const float* u1 = (const float*)d_in[3];
  const float* w2 = (const float*)d_in[4];
  const float* b2 = (const float*)d_in[5];
  const float* u2 = (const float*)d_in[6];
  const float* w3 = (const float*)d_in[7];
  const float* b3 = (const float*)d_in[8];
  const float* u3 = (const float*)d_in[9];
  float* out = (float*)d_out;

  float* ws    = (float*)d_ws;
  float* red   = ws;              // 32768: avg[16384] then max[16384] (stacked A)
  float* sbuf  = ws + 32768;      // 16384: channel attention s
  float* g2    = ws + 49152;      // 32768: GEMM2 output scratch
  float* ca    = ws + 81920;      // 131072
  float* cm    = ws + 212992;     // 131072
  float* sattn = ws + 344064;     // 131072
  float* invsv = ws + 475136;     // 3

  reduce_hw_kernel<<<16384, 256, 0, stream>>>(x, red, red + 16384);
  sn_kernel<<<1, 256, 0, stream>>>(w1, u1, w2, u2, w3, u3, invsv);
  mlp_wmma_kernel<<<1, 256, 0, stream>>>(red, w1, b1, w2, b2, invsv, g2, sbuf);
  dim3 gBP(16, 32);
  channel_reduce_kernel<<<gBP, 256, 0, stream>>>(x, sbuf, ca, cm);
  spatial_conv_kernel<<<gBP, 256, 0, stream>>>(ca, cm, w3, b3, invsv, sattn);
  final_mul_kernel<<<65536, 256, 0, stream>>>(x, sbuf, sattn, out);
}